// FTScanMultipleCarry_56298431316004
// MI455X (gfx1250) — compile-verified
//
#include <hip/hip_runtime.h>

// f32 scan over rows of (8192, 4096): s += x; p *= x; y = s + p.
// Bandwidth-bound (AI ~0.4 flop/byte). 3-pass chunked scan; MI455X 192MB L2
// makes the pass-3 re-read of xs ~free if ys stores are non-temporal.

typedef float v4f __attribute__((ext_vector_type(4)));

#define ROWS   8192
#define COLS   4096
#define COLW   (COLS / 4)        // 1024 vec4 columns
#define NCHUNK 128               // row chunks
#define CROWS  (ROWS / NCHUNK)   // 64 rows per chunk
#define TPB    256               // 8 wave32s per block
#define CTILES (COLW / TPB)      // 4 column tiles per chunk

// ---------------- Pass 1: per-chunk (sum, prod) aggregates ----------------
__global__ __launch_bounds__(TPB)
void FTScan_pass1(const float* __restrict__ xs,
                  float* __restrict__ pS, float* __restrict__ pP) {
    const int j4 = blockIdx.x * TPB + threadIdx.x;   // vec4 column 0..1023
    const int c  = blockIdx.y;                       // chunk 0..127
    const v4f* __restrict__ x4 = (const v4f*)xs;
    const int base = c * CROWS * COLW + j4;

    v4f s = {0.f, 0.f, 0.f, 0.f};
    v4f p = {1.f, 1.f, 1.f, 1.f};
    for (int r0 = 0; r0 < CROWS; r0 += 8) {
        if (r0 + 16 < CROWS)
            __builtin_prefetch(&x4[base + (r0 + 16) * COLW], 0, 1); // global_prefetch_b8
#pragma unroll
        for (int r = r0; r < r0 + 8; ++r) {
            v4f x = x4[base + r * COLW];   // global_load_b128, RT: keep xs in L2 for pass 3
            s += x;
            p *= x;
        }
    }
    ((v4f*)pS)[c * COLW + j4] = s;
    ((v4f*)pP)[c * COLW + j4] = p;
}

// ---- Pass 2: exclusive scan over chunk carries; emit final_sum/final_prod ----
__global__ __launch_bounds__(TPB)
void FTScan_pass2(float* __restrict__ pS, float* __restrict__ pP,
                  float* __restrict__ out) {
    const int j4 = blockIdx.x * TPB + threadIdx.x;   // vec4 column 0..1023
    v4f* S4 = (v4f*)pS;
    v4f* P4 = (v4f*)pP;
    v4f s = {0.f, 0.f, 0.f, 0.f};
    v4f p = {1.f, 1.f, 1.f, 1.f};
    for (int c = 0; c < NCHUNK; ++c) {
        v4f S = S4[c * COLW + j4];
        v4f P = P4[c * COLW + j4];
        S4[c * COLW + j4] = s;   // carry-in for chunk c (exclusive)
        P4[c * COLW + j4] = p;
        s += S;
        p *= P;
    }
    ((v4f*)out)[j4]          = s;   // final_sum  (4096 floats)
    ((v4f*)(out + COLS))[j4] = p;   // final_prod (4096 floats)
}

// ------- Pass 3: replay chunks with carry-in, stream ys non-temporally -------
__global__ __launch_bounds__(TPB)
void FTScan_pass3(const float* __restrict__ xs,
                  const float* __restrict__ pS, const float* __restrict__ pP,
                  float* __restrict__ out) {
    const int j4 = blockIdx.x * TPB + threadIdx.x;
    const int c  = blockIdx.y;
    const v4f* __restrict__ x4 = (const v4f*)xs;
    v4f* __restrict__ y4 = (v4f*)(out + 2 * COLS);   // ys after the two carries

    v4f s = ((const v4f*)pS)[c * COLW + j4];
    v4f p = ((const v4f*)pP)[c * COLW + j4];
    const int base = c * CROWS * COLW + j4;

    for (int r0 = 0; r0 < CROWS; r0 += 8) {
        if (r0 + 16 < CROWS)
            __builtin_prefetch(&x4[base + (r0 + 16) * COLW], 0, 1);
#pragma unroll
        for (int r = r0; r < r0 + 8; ++r) {
            const int idx = base + r * COLW;
            v4f x = __builtin_nontemporal_load(&x4[idx]);  // dead after use: NT load
            s += x;
            p *= x;
            __builtin_nontemporal_store(s + p, &y4[idx]);  // don't let ys evict xs from L2
        }
    }
}

// ------------- Fallback if d_ws is too small: naive column scan -------------
__global__ __launch_bounds__(TPB)
void FTScan_naive(const float* __restrict__ xs, float* __restrict__ out) {
    const int j4 = blockIdx.x * TPB + threadIdx.x;
    if (j4 >= COLW) return;
    const v4f* __restrict__ x4 = (const v4f*)xs;
    v4f* __restrict__ y4 = (v4f*)(out + 2 * COLS);
    v4f s = {0.f, 0.f, 0.f, 0.f};
    v4f p = {1.f, 1.f, 1.f, 1.f};
    for (int r = 0; r < ROWS; ++r) {
        v4f x = x4[r * COLW + j4];
        s += x;
        p *= x;
        __builtin_nontemporal_store(s + p, &y4[r * COLW + j4]);
    }
    ((v4f*)out)[j4]          = s;
    ((v4f*)(out + COLS))[j4] = p;
}

extern "C" void kernel_launch(void* const* d_in, const int* in_sizes, int n_in,
                              void* d_out, int out_size, void* d_ws, size_t ws_size,
                              hipStream_t stream) {
    const float* xs  = (const float*)d_in[0];
    float*       out = (float*)d_out;

    const size_t ws_needed = (size_t)NCHUNK * COLS * 2 * sizeof(float); // 4 MiB
    if (ws_size < ws_needed) {
        FTScan_naive<<<CTILES, TPB, 0, stream>>>(xs, out);
        return;
    }

    float* pS = (float*)d_ws;
    float* pP = pS + (size_t)NCHUNK * COLS;

    dim3 grid13(CTILES, NCHUNK);  // 512 blocks = 4096 wave32s
    FTScan_pass1<<<grid13, TPB, 0, stream>>>(xs, pS, pP);
    FTScan_pass2<<<CTILES, TPB, 0, stream>>>(pS, pP, out);
    FTScan_pass3<<<grid13, TPB, 0, stream>>>(xs, pS, pP, out);
}